// SybilGAT_63359357550979
// MI455X (gfx1250) — compile-verified
//
#include <hip/hip_runtime.h>
#include <hip/hip_bf16.h>
#include <math.h>

typedef __attribute__((ext_vector_type(2))) float v2f;
typedef __attribute__((ext_vector_type(8))) float v8f;

#define N_NODES 50000
#define N_EDGES 800000
#define E_TOT   (N_EDGES + N_NODES)   // reference appends self loops
#define NEG_SLOPE 0.2f

__device__ __forceinline__ float lrelu(float x) { return x > 0.f ? x : NEG_SLOPE * x; }

// ---------------- CSR build (edges grouped by dst) ----------------
__global__ void k_degree(const int* __restrict__ ei, int* __restrict__ deg) {
  int e = blockIdx.x * blockDim.x + threadIdx.x;
  if (e >= E_TOT) return;
  int dst = (e < N_EDGES) ? ei[N_EDGES + e] : (e - N_EDGES);
  atomicAdd(&deg[dst], 1);
}

__global__ void k_scan(const int* __restrict__ deg, int* __restrict__ rowptr,
                       int* __restrict__ cursor) {
  __shared__ int sh[1024];
  __shared__ int carry;
  int t = threadIdx.x;
  if (t == 0) carry = 0;
  __syncthreads();
  for (int base = 0; base < N_NODES; base += 1024) {
    int i = base + t;
    int v = (i < N_NODES) ? deg[i] : 0;
    sh[t] = v;
    __syncthreads();
    for (int off = 1; off < 1024; off <<= 1) {
      int add = (t >= off) ? sh[t - off] : 0;
      __syncthreads();
      sh[t] += add;
      __syncthreads();
    }
    if (i < N_NODES) {
      int rp = carry + sh[t] - v;   // exclusive prefix
      rowptr[i] = rp;
      cursor[i] = rp;
    }
    __syncthreads();
    if (t == 0) carry += sh[1023];
    __syncthreads();
  }
  if (t == 0) rowptr[N_NODES] = carry;
}

__global__ void k_scatter(const int* __restrict__ ei, int* __restrict__ cursor,
                          int* __restrict__ col) {
  int e = blockIdx.x * blockDim.x + threadIdx.x;
  if (e >= E_TOT) return;
  int src, dst;
  if (e < N_EDGES) { src = ei[e]; dst = ei[N_EDGES + e]; }
  else             { src = e - N_EDGES; dst = src; }
  int pos = atomicAdd(&cursor[dst], 1);
  col[pos] = src;
}

// ---------------- fp32 WMMA GEMM: C[M,N] = A[M,K] * B[K,N] ----------------
// Compile-time N,K (strength-reduced addressing). One wave computes a 5x1
// group of 16x16 C tiles (80 rows): each k-step loads B once and feeds 5
// independent V_WMMA_F32_16X16X4_F32 ops -> 5x B reuse, latency overlap.
// Requires M % 80 == 0 (50000 = 625*80).
template <int N, int K>
__global__ void k_gemm_wmma(const float* __restrict__ A, const float* __restrict__ B,
                            float* __restrict__ C, int M) {
  constexpr int MB = 5;
  int wave = (blockIdx.x * blockDim.x + threadIdx.x) >> 5;
  int lane = threadIdx.x & 31;
  constexpr int tilesN = N >> 4;
  int nGroups = (M / (16 * MB)) * tilesN;
  if (wave >= nGroups) return;
  int m0 = (wave / tilesN) * (16 * MB);
  int n0 = (wave % tilesN) << 4;
  int half = lane >> 4;      // 0: lanes 0-15, 1: lanes 16-31
  int lr   = lane & 15;
  // A 16x4 fragment: v0 = A[m=lr][kb], v1 = A[m=lr][kb+1], kb = k0 + 2*half
  const float* Ar[MB];
#pragma unroll
  for (int i = 0; i < MB; ++i)
    Ar[i] = A + (size_t)(m0 + i * 16 + lr) * K + 2 * half;
  // B 4x16 fragment: v0 = B[kb][n=lr], v1 = B[kb+1][n=lr]
  const float* Bcol = B + (size_t)2 * half * N + n0 + lr;
  v8f acc[MB] = {};
#pragma unroll 2
  for (int k0 = 0; k0 < K; k0 += 4) {
    v2f b;
    b.x = Bcol[(size_t)k0 * N];
    b.y = Bcol[(size_t)k0 * N + N];
#pragma unroll
    for (int i = 0; i < MB; ++i) {
      v2f a;
      a.x = Ar[i][k0];
      a.y = Ar[i][k0 + 1];
      acc[i] = __builtin_amdgcn_wmma_f32_16x16x4_f32(false, a, false, b,
                                                     (short)0, acc[i], false, false);
    }
  }
  // C/D 16x16: VGPR v holds row M = v + 8*half, col N = lr
#pragma unroll
  for (int i = 0; i < MB; ++i) {
    float* Crow = C + (size_t)(m0 + i * 16 + 8 * half) * N + n0 + lr;
#pragma unroll
    for (int v = 0; v < 8; ++v) Crow[(size_t)v * N] = acc[i][v];
  }
}

// ---------------- attention scores: one wave per (node, head), coalesced ----------------
__global__ void k_attn(const float* __restrict__ h, const float* __restrict__ att_s,
                       const float* __restrict__ att_d, float* __restrict__ asrc,
                       float* __restrict__ adst, int H, int C) {
  int wave = (blockIdx.x * blockDim.x + threadIdx.x) >> 5;
  int lane = threadIdx.x & 31;
  if (wave >= N_NODES * H) return;
  int n = wave / H, hh = wave % H;
  const float* hp = h + (size_t)n * H * C + (size_t)hh * C;
  const float* as = att_s + hh * C;
  const float* ad = att_d + hh * C;
  float s = 0.f, d = 0.f;
  for (int c = lane; c < C; c += 32) {
    float v = hp[c];
    s += v * as[c];
    d += v * ad[c];
  }
  for (int off = 16; off > 0; off >>= 1) {
    s += __shfl_xor(s, off, 32);
    d += __shfl_xor(d, off, 32);
  }
  if (lane == 0) { asrc[wave] = s; adst[wave] = d; }
}

// ---------------- segment softmax stats: one wave per (node, head) ----------------
__global__ void k_stats(const int* __restrict__ rowptr, const int* __restrict__ col,
                        const float* __restrict__ asrc, const float* __restrict__ adst,
                        float* __restrict__ mbuf, float* __restrict__ dinv, int H) {
  int wave = (blockIdx.x * blockDim.x + threadIdx.x) >> 5;
  int lane = threadIdx.x & 31;
  if (wave >= N_NODES * H) return;
  int n = wave / H, hh = wave % H;
  int beg = rowptr[n], end = rowptr[n + 1];
  float ad = adst[n * H + hh];
  float m = -__builtin_inff();
  for (int k = beg + lane; k < end; k += 32)
    m = fmaxf(m, lrelu(asrc[col[k] * H + hh] + ad));
  for (int off = 16; off > 0; off >>= 1) m = fmaxf(m, __shfl_xor(m, off, 32));
  float s = 0.f;
  for (int k = beg + lane; k < end; k += 32)
    s += __expf(lrelu(asrc[col[k] * H + hh] + ad) - m);
  for (int off = 16; off > 0; off >>= 1) s += __shfl_xor(s, off, 32);
  if (lane == 0) { mbuf[wave] = m; dinv[wave] = 1.f / (s + 1e-16f); }
}

// ---------------- aggregation + bias + ELU: one block (H*C threads) per node ----------------
// Per-(edge,head) softmax weights are computed once per chunk into LDS; the
// main loop is ds_load + coalesced global gather + FMA.
#define AGG_CHUNK 32
__global__ void k_aggregate(const int* __restrict__ rowptr, const int* __restrict__ col,
                            const float* __restrict__ asrc, const float* __restrict__ adst,
                            const float* __restrict__ mbuf, const float* __restrict__ dinv,
                            const float* __restrict__ h, const float* __restrict__ bias,
                            float* __restrict__ out, int H, int C) {
  __shared__ int   sh_src[AGG_CHUNK];
  __shared__ float sh_w[AGG_CHUNK * 4];   // max H = 4
  int n = blockIdx.x;
  int t = threadIdx.x;            // t in [0, H*C)
  int hh = t / C;
  int HC = H * C;
  int beg = rowptr[n], end = rowptr[n + 1];
  float acc = 0.f;
  for (int base = beg; base < end; base += AGG_CHUNK) {
    int cnt = min(AGG_CHUNK, end - base);
    __syncthreads();
    if (t < cnt * H) {
      int e  = t / H;
      int hw = t - e * H;
      int src = col[base + e];
      if (hw == 0) sh_src[e] = src;
      float ev = lrelu(asrc[src * H + hw] + adst[n * H + hw]);
      sh_w[e * H + hw] = __expf(ev - mbuf[n * H + hw]) * dinv[n * H + hw];
    }
    __syncthreads();
    for (int e = 0; e < cnt; ++e)
      acc += sh_w[e * H + hh] * h[(size_t)sh_src[e] * HC + t];
  }
  float o = acc + bias[t];
  out[(size_t)n * HC + t] = (o > 0.f) ? o : expm1f(o);   // ELU
}

// ---------------- layer-3 tiny GEMM (N=2) ----------------
__global__ void k_gemm3(const float* __restrict__ A, const float* __restrict__ W,
                        float* __restrict__ C) {
  int n = blockIdx.x * blockDim.x + threadIdx.x;
  if (n >= N_NODES) return;
  const float* a = A + (size_t)n * 128;
  float c0 = 0.f, c1 = 0.f;
  for (int k = 0; k < 128; ++k) {
    float v = a[k];
    c0 += v * W[2 * k];
    c1 += v * W[2 * k + 1];
  }
  C[2 * n] = c0;
  C[2 * n + 1] = c1;
}

// ---------------- layer-3 aggregation + bias + log_softmax ----------------
__global__ void k_final(const int* __restrict__ rowptr, const int* __restrict__ col,
                        const float* __restrict__ asrc, const float* __restrict__ adst,
                        const float* __restrict__ mbuf, const float* __restrict__ dinv,
                        const float* __restrict__ h3, const float* __restrict__ b3,
                        float* __restrict__ out) {
  int n = blockIdx.x * blockDim.x + threadIdx.x;
  if (n >= N_NODES) return;
  float ad = adst[n], m = mbuf[n], inv = dinv[n];
  float a0 = 0.f, a1 = 0.f;
  int beg = rowptr[n], end = rowptr[n + 1];
  for (int k = beg; k < end; ++k) {
    int src = col[k];
    float w = __expf(lrelu(asrc[src] + ad) - m) * inv;
    a0 += w * h3[2 * src];
    a1 += w * h3[2 * src + 1];
  }
  float v0 = a0 + b3[0], v1 = a1 + b3[1];
  float mm  = fmaxf(v0, v1);
  float lse = mm + logf(expf(v0 - mm) + expf(v1 - mm));
  out[2 * n]     = v0 - lse;
  out[2 * n + 1] = v1 - lse;
}

extern "C" void kernel_launch(void* const* d_in, const int* in_sizes, int n_in,
                              void* d_out, int out_size, void* d_ws, size_t ws_size,
                              hipStream_t stream) {
  const float* x   = (const float*)d_in[0];
  const int*   ei  = (const int*)d_in[1];
  const float* W1  = (const float*)d_in[2];
  const float* as1 = (const float*)d_in[3];
  const float* ad1 = (const float*)d_in[4];
  const float* b1  = (const float*)d_in[5];
  const float* W2  = (const float*)d_in[6];
  const float* as2 = (const float*)d_in[7];
  const float* ad2 = (const float*)d_in[8];
  const float* b2  = (const float*)d_in[9];
  const float* W3  = (const float*)d_in[10];
  const float* as3 = (const float*)d_in[11];
  const float* ad3 = (const float*)d_in[12];
  const float* b3  = (const float*)d_in[13];
  float* out = (float*)d_out;

  // Workspace layout (256B aligned); peak ~110 MB, L2-resident on MI455X.
  size_t off = 0;
  auto alloc = [&](size_t bytes) -> void* {
    void* p = (char*)d_ws + off;
    off += (bytes + 255) & ~(size_t)255;
    return p;
  };
  float* bufA  = (float*)alloc((size_t)N_NODES * 256 * 4); // h1 -> h2 -> h3
  float* bufB  = (float*)alloc((size_t)N_NODES * 256 * 4); // o1 -> o2
  float* asrc  = (float*)alloc((size_t)N_NODES * 4 * 4);
  float* adst  = (float*)alloc((size_t)N_NODES * 4 * 4);
  float* mbuf  = (float*)alloc((size_t)N_NODES * 4 * 4);
  float* dinv  = (float*)alloc((size_t)N_NODES * 4 * 4);
  int*   deg   = (int*)alloc((size_t)N_NODES * 4);
  int*   rowp  = (int*)alloc((size_t)(N_NODES + 1) * 4);
  int*   curs  = (int*)alloc((size_t)N_NODES * 4);
  int*   col   = (int*)alloc((size_t)E_TOT * 4);

  const int TB = 256;
  int ebl = (E_TOT + TB - 1) / TB;

  // ---- CSR build ----
  hipMemsetAsync(deg, 0, (size_t)N_NODES * 4, stream);
  k_degree<<<ebl, TB, 0, stream>>>(ei, deg);
  k_scan<<<1, 1024, 0, stream>>>(deg, rowp, curs);
  k_scatter<<<ebl, TB, 0, stream>>>(ei, curs, col);

  // ---- Layer 1: 128 -> 4 heads x 64 (concat) ----
  {
    const int H = 4, C = 64;
    int waves = (N_NODES / 80) * (256 / 16);
    k_gemm_wmma<256, 128><<<(waves * 32 + TB - 1) / TB, TB, 0, stream>>>(x, W1, bufA, N_NODES);
    k_attn<<<(N_NODES * H * 32 + TB - 1) / TB, TB, 0, stream>>>(bufA, as1, ad1, asrc, adst, H, C);
    k_stats<<<(N_NODES * H * 32 + TB - 1) / TB, TB, 0, stream>>>(rowp, col, asrc, adst, mbuf, dinv, H);
    k_aggregate<<<N_NODES, H * C, 0, stream>>>(rowp, col, asrc, adst, mbuf, dinv, bufA, b1, bufB, H, C);
  }

  // ---- Layer 2: 256 -> 2 heads x 64 (concat) ----
  {
    const int H = 2, C = 64;
    int waves = (N_NODES / 80) * (128 / 16);
    k_gemm_wmma<128, 256><<<(waves * 32 + TB - 1) / TB, TB, 0, stream>>>(bufB, W2, bufA, N_NODES);
    k_attn<<<(N_NODES * H * 32 + TB - 1) / TB, TB, 0, stream>>>(bufA, as2, ad2, asrc, adst, H, C);
    k_stats<<<(N_NODES * H * 32 + TB - 1) / TB, TB, 0, stream>>>(rowp, col, asrc, adst, mbuf, dinv, H);
    k_aggregate<<<N_NODES, H * C, 0, stream>>>(rowp, col, asrc, adst, mbuf, dinv, bufA, b2, bufB, H, C);
  }

  // ---- Layer 3: 128 -> 1 head x 2, mean(=identity), log_softmax ----
  {
    float* h3 = bufA;  // [N,2]
    k_gemm3<<<(N_NODES + TB - 1) / TB, TB, 0, stream>>>(bufB, W3, h3);
    k_attn<<<(N_NODES * 32 + TB - 1) / TB, TB, 0, stream>>>(h3, as3, ad3, asrc, adst, 1, 2);
    k_stats<<<(N_NODES * 32 + TB - 1) / TB, TB, 0, stream>>>(rowp, col, asrc, adst, mbuf, dinv, 1);
    k_final<<<(N_NODES + TB - 1) / TB, TB, 0, stream>>>(rowp, col, asrc, adst, mbuf, dinv, h3, b3, out);
  }
}